// PanoformerBlock_18038862643333
// MI455X (gfx1250) — compile-verified
//
#include <hip/hip_runtime.h>
#include <hip/hip_bf16.h>
#include <math.h>

// ---------------- problem constants ----------------
#define Bn      2
#define Hh      128
#define Ww      256
#define Cc      64
#define NHEAD   2
#define Kk      9
#define HIDDEN  256
#define HD      (Cc / NHEAD)          // 32 == wave32 lanes
#define NT      (Bn * Hh * Ww)        // 65536 tokens
#define HWi     (Hh * Ww)             // 32768

// ---------------- scratch arena layout (floats) ----------------
// total = 41,943,040 floats = 160 MiB
#define OFF_X1   ((size_t)0)                       // NT*64  : x + attn residual
#define OFF_HN   ((size_t)NT * 64)                 // NT*64  : LN output (h1 then h2)
#define OFF_P    ((size_t)2 * NT * 64)             // reused region
#define OFF_OFF  (OFF_P)                           // NT*36  : offsets
#define OFF_ATT  (OFF_OFF + (size_t)NT * 36)       // NT*18  : attn logits->softmax
#define OFF_V    (OFF_ATT + (size_t)NT * 18)       // NT*64  : value proj
#define OFF_OUT  (OFF_V   + (size_t)NT * 64)       // NT*64  : sampled attn out
#define OFF_Y1   (OFF_P)                           // NT*256 : LeFF hidden (phase B)
#define OFF_Y2   (OFF_Y1 + (size_t)NT * 256)       // NT*256 : dwconv out

typedef __attribute__((ext_vector_type(2))) float v2f;
typedef __attribute__((ext_vector_type(8))) float v8f;

__device__ __forceinline__ float gelu_exact(float x) {
    return 0.5f * x * (1.0f + erff(x * 0.70710678118654752440f));
}

// ---------------- LayerNorm: one wave32 per token (C=64 -> float2/lane) ----
__global__ void ln_kernel(const float* __restrict__ x,
                          const float* __restrict__ g,
                          const float* __restrict__ be,
                          float* __restrict__ out) {
    const int lane = threadIdx.x & 31;
    const int wid  = threadIdx.x >> 5;
    const int tok  = blockIdx.x * 8 + wid;
    if (tok >= NT) return;
    const float2 v = *(const float2*)(x + (size_t)tok * Cc + lane * 2);
    float s = v.x + v.y;
    #pragma unroll
    for (int o = 16; o >= 1; o >>= 1) s += __shfl_xor(s, o, 32);
    const float mean = s * (1.0f / Cc);
    const float d0 = v.x - mean, d1 = v.y - mean;
    float sq = d0 * d0 + d1 * d1;
    #pragma unroll
    for (int o = 16; o >= 1; o >>= 1) sq += __shfl_xor(sq, o, 32);
    const float inv = rsqrtf(sq * (1.0f / Cc) + 1e-5f);
    float2 r;
    r.x = d0 * inv * g[lane * 2 + 0] + be[lane * 2 + 0];
    r.y = d1 * inv * g[lane * 2 + 1] + be[lane * 2 + 1];
    *(float2*)(out + (size_t)tok * Cc + lane * 2) = r;
}

// ---------------- generic fp32 WMMA GEMM: out = A(MxK) @ W(KxN) + bias ------
// block = 256 threads = 8 waves; tile 32(M) x 64(N); K chunked by 64 via LDS.
// epilogue: +bias[col]; optional residual (res[M,N]); optional exact GELU.
__global__ void gemm_wmma_f32(const float* __restrict__ A,
                              const float* __restrict__ Wt,
                              const float* __restrict__ bias,
                              const float* __restrict__ res,
                              float* __restrict__ out,
                              int M, int K, int N, int do_gelu) {
    __shared__ float As[32][66];   // row stride even -> 8B-aligned float2 reads
    __shared__ float Bs[64][65];

    const int tid  = threadIdx.x;
    const int lane = tid & 31;
    const int wid  = tid >> 5;
    const int tr   = wid >> 2;         // 0..1 : wave row tile
    const int tc   = wid & 3;          // 0..3 : wave col tile
    const int row0 = blockIdx.x * 32;
    const int colB = blockIdx.y * 64;

    const int m  = lane & 15;
    const int n  = lane & 15;
    const int hi = lane >> 4;
    const int kk = hi * 2;

    v8f acc = {};

    for (int kc = 0; kc < K; kc += 64) {
        // stage A: 32 x 64 (coalesced in k)
        for (int idx = tid; idx < 32 * 64; idx += 256) {
            const int r = idx >> 6, c = idx & 63;
            As[r][c] = A[(size_t)(row0 + r) * K + kc + c];
        }
        // stage B: 64 x 64 (coalesced in n, guard col < N)
        for (int idx = tid; idx < 64 * 64; idx += 256) {
            const int r = idx >> 6, c = idx & 63;
            const int col = colB + c;
            Bs[r][c] = (col < N) ? Wt[(size_t)(kc + r) * N + col] : 0.0f;
        }
        __syncthreads();

        #pragma unroll
        for (int k = 0; k < 64; k += 4) {
            v2f a, b;
            a[0] = As[tr * 16 + m][k + kk];
            a[1] = As[tr * 16 + m][k + kk + 1];
            b[0] = Bs[k + kk][tc * 16 + n];
            b[1] = Bs[k + kk + 1][tc * 16 + n];
            acc = __builtin_amdgcn_wmma_f32_16x16x4_f32(
                      false, a, false, b, (short)0, acc, false, false);
        }
        __syncthreads();
    }

    const int colg = colB + tc * 16 + n;
    if (colg < N) {
        const float bv = bias[colg];
        #pragma unroll
        for (int r = 0; r < 8; ++r) {
            const int rowg = row0 + tr * 16 + r + 8 * hi;
            float v = acc[r] + bv;
            if (res) v += res[(size_t)rowg * N + colg];
            if (do_gelu) v = gelu_exact(v);
            out[(size_t)rowg * N + colg] = v;
        }
    }
}

// ---------------- softmax over K=9 per (token, head) ------------------------
__global__ void softmax9_kernel(float* __restrict__ att) {
    const int t = blockIdx.x * blockDim.x + threadIdx.x;
    if (t >= NT * NHEAD) return;
    float* p = att + (size_t)(t >> 1) * (NHEAD * Kk) + (t & 1) * Kk;
    float mx = p[0];
    #pragma unroll
    for (int k = 1; k < Kk; ++k) mx = fmaxf(mx, p[k]);
    float s = 0.0f, e[Kk];
    #pragma unroll
    for (int k = 0; k < Kk; ++k) { e[k] = __expf(p[k] - mx); s += e[k]; }
    const float inv = 1.0f / s;
    #pragma unroll
    for (int k = 0; k < Kk; ++k) p[k] = e[k] * inv;
}

// ---------------- bilinear grid-sample + attention weighting ----------------
// one wave32 per (token, head); lane == channel (HD==32). Coalesced gathers.
__global__ void sample_kernel(const float* __restrict__ ref,
                              const float* __restrict__ off,
                              const float* __restrict__ att,
                              const float* __restrict__ vbuf,
                              float* __restrict__ outs) {
    const int gtid = blockIdx.x * 256 + threadIdx.x;
    const int gw   = gtid >> 5;                  // wave id
    const int lane = threadIdx.x & 31;
    if (gw >= NT * NHEAD) return;
    const int tok  = gw >> 1;
    const int head = gw & 1;
    const int b  = tok >> 15;                    // / HWi
    const int hw = tok & (HWi - 1);
    const size_t rbase = (size_t)tok * (Kk * 2); // ref (B,H,W,K,2) == tok*18
    const size_t obase = (size_t)tok * (NHEAD * Kk * 2);
    const size_t abase = (size_t)tok * (NHEAD * Kk) + head * Kk;
    const size_t vchan = (size_t)head * HD + lane;

    float acc = 0.0f;
    #pragma unroll
    for (int k = 0; k < Kk; ++k) {
        const float gx = ref[rbase + k * 2 + 0] + off[obase + (head * Kk + k) * 2 + 0];
        const float gy = ref[rbase + k * 2 + 1] + off[obase + (head * Kk + k) * 2 + 1];
        const float aw = att[abase + k];
        const float ix = (gx + 1.0f) * (Ww * 0.5f) - 0.5f;
        const float iy = (gy + 1.0f) * (Hh * 0.5f) - 0.5f;
        const float x0 = floorf(ix), y0 = floorf(iy);
        const float wx = ix - x0,    wy = iy - y0;

        float cs[4];
        #pragma unroll
        for (int c = 0; c < 4; ++c) {
            const float xf = x0 + (float)(c & 1);
            const float yf = y0 + (float)(c >> 1);
            const bool ok = (xf >= 0.0f) && (xf <= (float)(Ww - 1)) &&
                            (yf >= 0.0f) && (yf <= (float)(Hh - 1));
            const int xc = (int)fminf(fmaxf(xf, 0.0f), (float)(Ww - 1));
            const int yc = (int)fminf(fmaxf(yf, 0.0f), (float)(Hh - 1));
            const float val = vbuf[((size_t)b * HWi + (size_t)yc * Ww + xc) * Cc + vchan];
            cs[c] = ok ? val : 0.0f;
        }
        acc += aw * (cs[0] * (1.0f - wx) * (1.0f - wy) + cs[1] * wx * (1.0f - wy) +
                     cs[2] * (1.0f - wx) * wy          + cs[3] * wx * wy);
    }
    outs[(size_t)tok * Cc + vchan] = acc;
}

// ---------------- depthwise 3x3, circular-W / zero-H pad, + GELU ------------
__global__ void dwconv_kernel(const float* __restrict__ y1,
                              const float* __restrict__ Wdw,
                              const float* __restrict__ bdw,
                              float* __restrict__ y2) {
    const size_t t = (size_t)blockIdx.x * 256 + threadIdx.x;
    if (t >= (size_t)NT * HIDDEN) return;
    const int c   = (int)(t & (HIDDEN - 1));
    const int tok = (int)(t >> 8);
    const int b = tok >> 15;
    const int h = (tok & (HWi - 1)) >> 8;
    const int w = tok & (Ww - 1);

    float acc = bdw[c];
    #pragma unroll
    for (int dy = -1; dy <= 1; ++dy) {
        const int hh = h + dy;
        if (hh < 0 || hh >= Hh) continue;
        #pragma unroll
        for (int dx = -1; dx <= 1; ++dx) {
            const int ww = (w + dx + Ww) & (Ww - 1);   // circular in width
            const float pix = y1[(((size_t)b * Hh + hh) * Ww + ww) * HIDDEN + c];
            acc += pix * Wdw[c * 9 + (dy + 1) * 3 + (dx + 1)];
        }
    }
    y2[t] = gelu_exact(acc);
}

// ---------------- host-side launcher ----------------------------------------
extern "C" void kernel_launch(void* const* d_in, const int* in_sizes, int n_in,
                              void* d_out, int out_size, void* d_ws, size_t ws_size,
                              hipStream_t stream) {
    const float* x    = (const float*)d_in[0];
    const float* ref  = (const float*)d_in[1];
    const float* g1   = (const float*)d_in[2];
    const float* be1  = (const float*)d_in[3];
    const float* g2   = (const float*)d_in[4];
    const float* be2  = (const float*)d_in[5];
    const float* Woff = (const float*)d_in[6];
    const float* boff = (const float*)d_in[7];
    const float* Wa   = (const float*)d_in[8];
    const float* ba   = (const float*)d_in[9];
    const float* Wv   = (const float*)d_in[10];
    const float* bv   = (const float*)d_in[11];
    const float* Wo   = (const float*)d_in[12];
    const float* bo   = (const float*)d_in[13];
    const float* W1   = (const float*)d_in[14];
    const float* c1   = (const float*)d_in[15];
    const float* Wdw  = (const float*)d_in[16];
    const float* bdw  = (const float*)d_in[17];
    const float* W2   = (const float*)d_in[18];
    const float* c2   = (const float*)d_in[19];
    float* out = (float*)d_out;
    float* ws  = (float*)d_ws;

    float* h    = ws + OFF_HN;
    float* offb = ws + OFF_OFF;
    float* attb = ws + OFF_ATT;
    float* vb   = ws + OFF_V;
    float* outs = ws + OFF_OUT;
    float* x1   = ws + OFF_X1;
    float* y1   = ws + OFF_Y1;
    float* y2   = ws + OFF_Y2;

    const dim3 blk(256);

    // 1) LN1
    ln_kernel<<<NT / 8, blk, 0, stream>>>(x, g1, be1, h);
    // 2) projections (WMMA fp32)
    gemm_wmma_f32<<<dim3(NT / 32, 1), blk, 0, stream>>>(h, Woff, boff, nullptr, offb,
                                                        NT, Cc, NHEAD * Kk * 2, 0);
    gemm_wmma_f32<<<dim3(NT / 32, 1), blk, 0, stream>>>(h, Wa, ba, nullptr, attb,
                                                        NT, Cc, NHEAD * Kk, 0);
    gemm_wmma_f32<<<dim3(NT / 32, 1), blk, 0, stream>>>(h, Wv, bv, nullptr, vb,
                                                        NT, Cc, Cc, 0);
    // 3) softmax over K=9
    softmax9_kernel<<<(NT * NHEAD + 255) / 256, blk, 0, stream>>>(attb);
    // 4) deformable bilinear sampling + weighting
    sample_kernel<<<(NT * NHEAD) / 8, blk, 0, stream>>>(ref, offb, attb, vb, outs);
    // 5) output proj + residual -> x1
    gemm_wmma_f32<<<dim3(NT / 32, 1), blk, 0, stream>>>(outs, Wo, bo, x, x1,
                                                        NT, Cc, Cc, 0);
    // 6) LN2 (reuse h)
    ln_kernel<<<NT / 8, blk, 0, stream>>>(x1, g2, be2, h);
    // 7) LeFF up-proj + GELU
    gemm_wmma_f32<<<dim3(NT / 32, 4), blk, 0, stream>>>(h, W1, c1, nullptr, y1,
                                                        NT, Cc, HIDDEN, 1);
    // 8) depthwise conv + GELU
    dwconv_kernel<<<(unsigned)(((size_t)NT * HIDDEN) / 256), blk, 0, stream>>>(y1, Wdw, bdw, y2);
    // 9) LeFF down-proj + residual -> out
    gemm_wmma_f32<<<dim3(NT / 32, 1), blk, 0, stream>>>(y2, W2, c2, x1, out,
                                                        NT, HIDDEN, Cc, 0);
}